// TotalPairwiseConfusionLoss_77936476553597
// MI455X (gfx1250) — compile-verified
//
#include <hip/hip_runtime.h>
#include <hip/hip_bf16.h>

typedef __attribute__((ext_vector_type(2))) float v2f;
typedef __attribute__((ext_vector_type(8))) float v8f;

#define TPC_ROWS 256
#define TPC_TILE 16
#define TPC_COLS 100000   // reference shape: x is (256, 100000) fp32

// Pass 1: one wave (block of 32) per 16-column tile.
// A = ones(16x4); D = A*B + C accumulates column sums of each 4x16 B tile.
// acc_cs accumulates column sums, acc_sq accumulates column sums-of-squares.
// SCOLS != 0 -> compile-time column stride (immediate load offsets, no per-load
// address math); SCOLS == 0 -> generic runtime-stride fallback.
template <int SCOLS>
__global__ __launch_bounds__(32) void tpc_pass1(const float* __restrict__ x,
                                                float* __restrict__ partials,
                                                int cols_arg) {
    const int cols  = SCOLS ? SCOLS : cols_arg;
    const int lane  = threadIdx.x & 31;
    const int n     = lane & 15;              // column within tile (B/D: N = lane%16)
    const int kHalf = (lane >> 4) << 1;       // lanes 0-15 -> rows k+0,k+1 ; 16-31 -> k+2,k+3

    // Base pointer already includes this lane's column and its K-half row offset.
    const float* p = x + (size_t)blockIdx.x * TPC_TILE + (size_t)n
                       + (size_t)kHalf * (size_t)cols;

    v2f a; a.x = 1.0f; a.y = 1.0f;            // all-ones A matrix (16x4)
    v8f acc_cs = {};                          // column sums
    v8f acc_sq = {};                          // column sums of squares

    for (int r = 0; r < TPC_ROWS; r += 16) {
        // Prefetch the next 16-row block (speculative; dropped past the end).
        __builtin_prefetch(p + (size_t)16 * (size_t)cols, 0, 1);

        #pragma unroll
        for (int g = 0; g < 4; ++g) {
            // Offsets are compile-time when SCOLS != 0: {0,1,4,5,8,9,12,13}*COLS*4B,
            // all within the 24-bit signed immediate range.
            const float e0 = p[(size_t)(4 * g)     * (size_t)cols];
            const float e1 = p[(size_t)(4 * g + 1) * (size_t)cols];

            v2f b;  b.x  = e0;       b.y  = e1;
            v2f bq; bq.x = e0 * e0;  bq.y = e1 * e1;

            // 8 args: (neg_a, A, neg_b, B, c_mod, C, reuse_a, reuse_b)
            acc_cs = __builtin_amdgcn_wmma_f32_16x16x4_f32(
                false, a, false, b,  (short)0, acc_cs, false, false);
            acc_sq = __builtin_amdgcn_wmma_f32_16x16x4_f32(
                false, a, false, bq, (short)0, acc_sq, false, false);
        }
        p += (size_t)16 * (size_t)cols;       // single 64-bit pointer bump per block
    }

    // D VGPR0: lane L holds (M = L<16?0:8, N = L%16); all rows identical since A==1.
    const float cs  = acc_cs[0];              // full column sum for this lane's column
    float sq  = acc_sq[0];                    // full sum of squares for that column
    float cs2 = cs * cs;

    // Full-wave reduction; lanes 16-31 duplicate lanes 0-15, so halve at the end.
    #pragma unroll
    for (int m = 16; m >= 1; m >>= 1) {
        sq  += __shfl_xor(sq,  m, 32);
        cs2 += __shfl_xor(cs2, m, 32);
    }
    if (lane == 0) {
        partials[2 * blockIdx.x + 0] = sq  * 0.5f;
        partials[2 * blockIdx.x + 1] = cs2 * 0.5f;
    }
}

// Pass 2: deterministic final reduction of per-tile partials.
__global__ __launch_bounds__(256) void tpc_pass2(const float* __restrict__ partials,
                                                 int nblocks,
                                                 float* __restrict__ out,
                                                 float nrows) {
    __shared__ float s_sq[256];
    __shared__ float s_cs[256];
    float sq = 0.0f, cs2 = 0.0f;
    for (int i = threadIdx.x; i < nblocks; i += 256) {
        sq  += partials[2 * i + 0];
        cs2 += partials[2 * i + 1];
    }
    s_sq[threadIdx.x] = sq;
    s_cs[threadIdx.x] = cs2;
    __syncthreads();
    for (int s = 128; s > 0; s >>= 1) {
        if (threadIdx.x < s) {
            s_sq[threadIdx.x] += s_sq[threadIdx.x + s];
            s_cs[threadIdx.x] += s_cs[threadIdx.x + s];
        }
        __syncthreads();
    }
    if (threadIdx.x == 0) {
        out[0] = nrows * s_sq[0] - s_cs[0];
    }
}

extern "C" void kernel_launch(void* const* d_in, const int* in_sizes, int n_in,
                              void* d_out, int out_size, void* d_ws, size_t ws_size,
                              hipStream_t stream) {
    const float* x   = (const float*)d_in[0];
    float*       out = (float*)d_out;
    float*       partials = (float*)d_ws;     // 2 floats per tile (50 KB total)

    const int rows   = TPC_ROWS;              // reference: x is (256, 100000) fp32
    const int cols   = in_sizes[0] / rows;
    const int ntiles = cols / TPC_TILE;       // 6250 single-wave blocks

    if (cols == TPC_COLS) {
        tpc_pass1<TPC_COLS><<<ntiles, 32, 0, stream>>>(x, partials, cols);
    } else {
        tpc_pass1<0><<<ntiles, 32, 0, stream>>>(x, partials, cols);
    }
    tpc_pass2<<<1, 256, 0, stream>>>(partials, ntiles, out, (float)rows);
}